// PotentialLoss_46540265619936
// MI455X (gfx1250) — compile-verified
//
#include <hip/hip_runtime.h>
#include <hip/hip_bf16.h>
#include <math.h>

// Problem constants (fixed by the reference).
#define N_HITS   100000
#define P_PART   300
#define P_PAD    320                 // pad particles to multiple of 16; padded cols have q_a=0
#define TILES_N  (P_PAD / 16)        // 20
#define TILES_M  (N_HITS / 16)       // 6250 (N divisible by 16)
#define WAVES_PER_BLOCK 8            // 256 threads = 8 wave32
#define K3_BLOCKS ((TILES_M * TILES_N) / WAVES_PER_BLOCK)  // 15625 exactly
#define Q_MIN    0.01f
#define RADIUS   10.0f
#define PT_THLD  0.9f

typedef __attribute__((ext_vector_type(2))) float v2f;
typedef __attribute__((ext_vector_type(8))) float v8f;

// ---- workspace layout (bytes) ----
#define WS_Q        0u                         // N floats: q
#define WS_XN       400000u                    // N floats: ||x||^2
#define WS_QMASK    800000u                    // N floats: q * maskf
#define WS_KEYS     1200000u                   // P u64 argmax keys (8-aligned)
#define WS_XA       1202400u                   // P_PAD*4 floats: centers
#define WS_QA       1207520u                   // P_PAD floats: q_a (0 in padding)
#define WS_XAN      1208800u                   // P_PAD floats: ||x_a||^2
#define WS_MCNT     1210080u                   // int mask count (pad to 8)
#define WS_PART     1210088u                   // K3_BLOCKS*2 floats partials

__global__ void init_kernel(unsigned long long* keys, int* mask_count) {
    int t = threadIdx.x;
    if (t < P_PART) keys[t] = 0ull;
    if (t == 0) *mask_count = 0;
}

__global__ void __launch_bounds__(256) hits_kernel(
    const float* __restrict__ beta, const float* __restrict__ x,
    const int* __restrict__ pid, const int* __restrict__ recon,
    const float* __restrict__ tp,
    float* __restrict__ q, float* __restrict__ xn, float* __restrict__ qmask,
    unsigned long long* __restrict__ keys, int* __restrict__ mask_count) {
    __shared__ int scnt[256];
    int i = blockIdx.x * 256 + threadIdx.x;
    int c = 0;
    if (i < N_HITS) {
        float b  = beta[i];
        float t  = atanhf(b);
        float qi = t * t + Q_MIN;
        q[i] = qi;
        float x0 = x[i * 4 + 0], x1 = x[i * 4 + 1];
        float x2 = x[i * 4 + 2], x3 = x[i * 4 + 3];
        xn[i] = x0 * x0 + x1 * x1 + x2 * x2 + x3 * x3;
        bool m = (recon[i] > 0) && (tp[i] > PT_THLD);
        qmask[i] = m ? qi : 0.0f;
        c = m ? 1 : 0;
        int p = pid[i];
        if (p > 0) {
            // larger q wins; on exact tie, smaller index wins (matches argmax)
            unsigned long long key =
                ((unsigned long long)__float_as_uint(qi) << 32) |
                (unsigned long long)(0xFFFFFFFFu - (unsigned)i);
            atomicMax(&keys[p - 1], key);
        }
    }
    scnt[threadIdx.x] = c;
    __syncthreads();
    for (int s = 128; s > 0; s >>= 1) {
        if (threadIdx.x < s) scnt[threadIdx.x] += scnt[threadIdx.x + s];
        __syncthreads();
    }
    if (threadIdx.x == 0 && scnt[0]) atomicAdd(mask_count, scnt[0]);
}

__global__ void gather_kernel(const float* __restrict__ x,
                              const float* __restrict__ q,
                              const float* __restrict__ xn,
                              const unsigned long long* __restrict__ keys,
                              float* __restrict__ xa, float* __restrict__ qa,
                              float* __restrict__ xan) {
    int n = threadIdx.x;
    if (n >= P_PAD) return;
    if (n < P_PART) {
        unsigned long long k = keys[n];
        unsigned idx = 0xFFFFFFFFu - (unsigned)(k & 0xFFFFFFFFull);
        xa[n * 4 + 0] = x[idx * 4 + 0];
        xa[n * 4 + 1] = x[idx * 4 + 1];
        xa[n * 4 + 2] = x[idx * 4 + 2];
        xa[n * 4 + 3] = x[idx * 4 + 3];
        qa[n]  = q[idx];
        xan[n] = xn[idx];
    } else {
        xa[n * 4 + 0] = 0.0f; xa[n * 4 + 1] = 0.0f;
        xa[n * 4 + 2] = 0.0f; xa[n * 4 + 3] = 0.0f;
        qa[n]  = 0.0f;   // padded columns contribute exactly 0 to both terms
        xan[n] = 0.0f;
    }
}

// One wave computes one 16(hits)x16(pids) tile:
//   dot tile via v_wmma_f32_16x16x4_f32, then a fully branchless epilogue.
__global__ void __launch_bounds__(256) pair_kernel(
    const float* __restrict__ x, const int* __restrict__ pid,
    const float* __restrict__ q, const float* __restrict__ xn,
    const float* __restrict__ qmask,
    const float* __restrict__ xa, const float* __restrict__ qa,
    const float* __restrict__ xan,
    float* __restrict__ partials) {
    __shared__ float sva[256];
    __shared__ float svr[256];
    const int tid  = threadIdx.x;
    const int lane = tid & 31;
    const int wave = tid >> 5;
    const int gtile = blockIdx.x * WAVES_PER_BLOCK + wave;  // always < TILES_M*TILES_N
    const int tm = gtile / TILES_N;
    const int tn = gtile % TILES_N;
    const int l16 = lane & 15;
    const int kh  = lane >> 4;       // K half: lanes<16 -> K{0,1}, lanes>=16 -> K{2,3}

    // A: 16x4 f32 (rows = hits). ISA layout: lane L (M=L%16), v0=K(2*kh), v1=K(2*kh+1)
    const int row = tm * 16 + l16;
    v2f a;
    a.x = x[row * 4 + 2 * kh];
    a.y = x[row * 4 + 2 * kh + 1];
    // B: 4x16 f32 (cols = particles), mirrored layout.
    const int col = tn * 16 + l16;
    v2f b;
    b.x = xa[col * 4 + 2 * kh];
    b.y = xa[col * 4 + 2 * kh + 1];

    v8f c = {};
    // D = A x B : D[v] holds (M = v + 8*kh, N = lane&15) per ISA C/D layout.
    v8f d = __builtin_amdgcn_wmma_f32_16x16x4_f32(
        /*neg_a=*/false, a, /*neg_b=*/false, b,
        /*c_mod=*/(short)0, c, /*reuse_a=*/false, /*reuse_b=*/false);

    const int   ncol1 = col + 1;     // pid value for this column
    const float qan   = qa[col];
    const float xann  = xan[col];
    const int   mbase = tm * 16 + 8 * kh;   // multiple of 8 -> 32B aligned

    // Vector loads of the 8 consecutive per-row values this lane needs.
    const int f4 = mbase >> 2;
    const float4* q4  = reinterpret_cast<const float4*>(q);
    const float4* xn4 = reinterpret_cast<const float4*>(xn);
    const float4* qm4 = reinterpret_cast<const float4*>(qmask);
    const int4*   pd4 = reinterpret_cast<const int4*>(pid);
    const float4 qv0  = q4[f4],  qv1  = q4[f4 + 1];
    const float4 xnv0 = xn4[f4], xnv1 = xn4[f4 + 1];
    const float4 qmv0 = qm4[f4], qmv1 = qm4[f4 + 1];
    const int4   pv0  = pd4[f4], pv1  = pd4[f4 + 1];

    const float qmArr[8]  = {qv0.x, qv0.y, qv0.z, qv0.w, qv1.x, qv1.y, qv1.z, qv1.w};
    const float xnArr[8]  = {xnv0.x, xnv0.y, xnv0.z, xnv0.w, xnv1.x, xnv1.y, xnv1.z, xnv1.w};
    const float qmkArr[8] = {qmv0.x, qmv0.y, qmv0.z, qmv0.w, qmv1.x, qmv1.y, qmv1.z, qmv1.w};
    const int   pArr[8]   = {pv0.x, pv0.y, pv0.z, pv0.w, pv1.x, pv1.y, pv1.z, pv1.w};

    float vaacc = 0.0f, vracc = 0.0f;
#pragma unroll
    for (int v = 0; v < 8; ++v) {
        float nsq = xnArr[v] + xann - 2.0f * d[v];
        nsq = fmaxf(nsq, 0.0f);                          // guard cancellation
        const float rep = fmaxf(RADIUS - __builtin_amdgcn_sqrtf(nsq + 1e-8f), 0.0f);
        const bool  same = (pArr[v] == ncol1);
        const float wa = same ? qmkArr[v] : 0.0f;        // attractive weight (masked q)
        const float wr = same ? 0.0f : qmArr[v];         // repulsive weight (q)
        vaacc = fmaf(wa * qan, nsq, vaacc);
        vracc = fmaf(wr * qan, rep, vracc);
    }
    sva[tid] = vaacc;
    svr[tid] = vracc;
    __syncthreads();
    for (int s = 128; s > 0; s >>= 1) {
        if (tid < s) { sva[tid] += sva[tid + s]; svr[tid] += svr[tid + s]; }
        __syncthreads();
    }
    if (tid == 0) {
        partials[blockIdx.x * 2 + 0] = sva[0];
        partials[blockIdx.x * 2 + 1] = svr[0];
    }
}

__global__ void __launch_bounds__(256) final_kernel(
    const float* __restrict__ partials, const int* __restrict__ mask_count,
    float* __restrict__ out) {
    __shared__ float sva[256];
    __shared__ float svr[256];
    float va = 0.0f, vr = 0.0f;
    for (int i = threadIdx.x; i < K3_BLOCKS; i += 256) {
        va += partials[i * 2 + 0];
        vr += partials[i * 2 + 1];
    }
    sva[threadIdx.x] = va;
    svr[threadIdx.x] = vr;
    __syncthreads();
    for (int s = 128; s > 0; s >>= 1) {
        if (threadIdx.x < s) {
            sva[threadIdx.x] += sva[threadIdx.x + s];
            svr[threadIdx.x] += svr[threadIdx.x + s];
        }
        __syncthreads();
    }
    if (threadIdx.x == 0) {
        const float mc = (float)(*mask_count);
        out[0] = sva[0] / mc;                 // attractive: masked column-mean sum
        out[1] = svr[0] / (float)N_HITS;      // repulsive: mean over hits, sum over pids
    }
}

extern "C" void kernel_launch(void* const* d_in, const int* in_sizes, int n_in,
                              void* d_out, int out_size, void* d_ws, size_t ws_size,
                              hipStream_t stream) {
    const float* beta  = (const float*)d_in[0];
    const float* x     = (const float*)d_in[1];
    const int*   pid   = (const int*)d_in[2];
    const int*   recon = (const int*)d_in[3];
    const float* tp    = (const float*)d_in[4];
    float* out = (float*)d_out;

    char* ws = (char*)d_ws;
    float* q      = (float*)(ws + WS_Q);
    float* xn     = (float*)(ws + WS_XN);
    float* qmask  = (float*)(ws + WS_QMASK);
    unsigned long long* keys = (unsigned long long*)(ws + WS_KEYS);
    float* xa     = (float*)(ws + WS_XA);
    float* qa     = (float*)(ws + WS_QA);
    float* xan    = (float*)(ws + WS_XAN);
    int*   mcnt   = (int*)(ws + WS_MCNT);
    float* part   = (float*)(ws + WS_PART);

    init_kernel<<<1, 512, 0, stream>>>(keys, mcnt);
    hits_kernel<<<(N_HITS + 255) / 256, 256, 0, stream>>>(
        beta, x, pid, recon, tp, q, xn, qmask, keys, mcnt);
    gather_kernel<<<1, P_PAD, 0, stream>>>(x, q, xn, keys, xa, qa, xan);
    pair_kernel<<<K3_BLOCKS, 256, 0, stream>>>(
        x, pid, q, xn, qmask, xa, qa, xan, part);
    final_kernel<<<1, 256, 0, stream>>>(part, mcnt, out);
}